// SAGEEncoder_16140487099036
// MI455X (gfx1250) — compile-verified
//
#include <hip/hip_runtime.h>

typedef float v2f __attribute__((ext_vector_type(2)));
typedef float v8f __attribute__((ext_vector_type(8)));

#define CH 64          // IN_CH == HID == 64
#define NGRAPH 64

// ---------------------------------------------------------------- zero fill
__global__ void zero_kernel(float* __restrict__ p, long n) {
  long i = (long)blockIdx.x * blockDim.x + threadIdx.x;
  long stride = (long)gridDim.x * blockDim.x;
  for (; i < n; i += stride) p[i] = 0.0f;
}

// ------------------------------------------------- edge scatter (mean-agg)
// msg[dst][c] += feat[src][c];  deg[dst] += 1
__global__ void scatter_kernel(const float* __restrict__ feat,
                               const int* __restrict__ src,
                               const int* __restrict__ dst,
                               float* __restrict__ msg,
                               float* __restrict__ deg,
                               long n_edges) {
  long tid = (long)blockIdx.x * blockDim.x + threadIdx.x;
  long stride = (long)gridDim.x * blockDim.x;
  long total = n_edges * CH;
  for (; tid < total; tid += stride) {
    long e = tid >> 6;
    int  c = (int)(tid & 63);
    int  s = src[e];
    int  d = dst[e];
    atomicAdd(&msg[(long)d * CH + c], feat[(long)s * CH + c]);
    if (c == 0) atomicAdd(&deg[d], 1.0f);
  }
}

// ------------------------------------------------------- fused SAGE GEMM
// out[16-node tile] = relu( [msg/deg | xin] (16x128) @ [w_l ; w_r] (128x64) + b )
// 128 threads = 4 waves; wave w computes output columns [16w, 16w+16).
__global__ __launch_bounds__(128)
void sage_gemm_kernel(const float* __restrict__ msg,
                      const float* __restrict__ deg,
                      const float* __restrict__ xin,
                      const float* __restrict__ w_l,
                      const float* __restrict__ w_r,
                      const float* __restrict__ bias,
                      float* __restrict__ out,
                      long n_nodes, int n_tiles) {
  __shared__ float Bs[128][CH + 4];   // [K][N], padded: bank = (4k + n) % 64
  __shared__ float As[16][128 + 4];   // [M][K], padded: bank = (4m + k) % 64

  const int t     = threadIdx.x;
  const int wave  = t >> 5;
  const int lane  = t & 31;
  const int lhalf = lane >> 4;   // 0: lanes 0-15, 1: lanes 16-31
  const int lmod  = lane & 15;

  // Stage combined weight matrix [w_l ; w_r] once per block.
  for (int i = t; i < 128 * CH; i += 128) {
    int k = i >> 6, n = i & 63;
    Bs[k][n] = (k < CH) ? w_l[k * CH + n] : w_r[(k - CH) * CH + n];
  }

  for (int tile = blockIdx.x; tile < n_tiles; tile += gridDim.x) {
    long base = (long)tile * 16;

    // Stage A: row m = t&15, 16-column chunk cb = t>>4 (0..7).
    {
      int  m    = t & 15;
      int  cb   = t >> 4;
      long node = base + m;
      if (node >= n_nodes) node = n_nodes - 1;   // clamp (N=100000 is exact anyway)
      if (cb < 4) {
        float rd = 1.0f / fmaxf(deg[node], 1.0f);
        const float* mrow = msg + node * CH + cb * 16;
        #pragma unroll
        for (int j = 0; j < 16; ++j) As[m][cb * 16 + j] = mrow[j] * rd;
      } else {
        const float* xrow = xin + node * CH + (cb - 4) * 16;
        #pragma unroll
        for (int j = 0; j < 16; ++j) As[m][CH + (cb - 4) * 16 + j] = xrow[j];
      }
    }
    __syncthreads();

    // K loop: 32 x v_wmma_f32_16x16x4_f32 per wave.
    v8f acc = {};
    #pragma unroll
    for (int k = 0; k < 128; k += 4) {
      v2f a, b;
      // A 16x4 layout: lanes 0-15 -> K = k+0 (v0), k+1 (v1); lanes 16-31 -> k+2, k+3
      a.x = As[lmod][k + lhalf * 2 + 0];
      a.y = As[lmod][k + lhalf * 2 + 1];
      // B 4x16 layout: N across lanes; lane halves split the K range likewise
      b.x = Bs[k + lhalf * 2 + 0][wave * 16 + lmod];
      b.y = Bs[k + lhalf * 2 + 1][wave * 16 + lmod];
      acc = __builtin_amdgcn_wmma_f32_16x16x4_f32(
          /*neg_a=*/false, a, /*neg_b=*/false, b,
          /*c_mod=*/(short)0, acc, /*reuse_a=*/false, /*reuse_b=*/false);
    }
    __syncthreads();   // before next tile overwrites As

    // C/D layout: VGPR r -> row (r + 8*lhalf), col = lane%16 (+ wave tile offset)
    int   col = wave * 16 + lmod;
    float bv  = bias[col];
    #pragma unroll
    for (int r = 0; r < 8; ++r) {
      int  m    = r + lhalf * 8;
      long node = base + m;
      if (node < n_nodes) {
        float v = acc[r] + bv;
        out[node * CH + col] = fmaxf(v, 0.0f);
      }
    }
  }
}

// ------------------------------------------------------- global mean pool
__global__ void pool_accum_kernel(const float* __restrict__ h,
                                  const int* __restrict__ batch,
                                  float* __restrict__ sums,
                                  float* __restrict__ counts,
                                  long n_nodes) {
  long tid = (long)blockIdx.x * blockDim.x + threadIdx.x;
  long stride = (long)gridDim.x * blockDim.x;
  long total = n_nodes * CH;
  for (; tid < total; tid += stride) {
    long n = tid >> 6;
    int  c = (int)(tid & 63);
    int  g = batch[n];
    atomicAdd(&sums[(long)g * CH + c], h[n * CH + c]);
    if (c == 0) atomicAdd(&counts[g], 1.0f);
  }
}

__global__ void pool_final_kernel(const float* __restrict__ sums,
                                  const float* __restrict__ counts,
                                  float* __restrict__ out) {
  int i = blockIdx.x * blockDim.x + threadIdx.x;   // 4096 threads
  if (i < NGRAPH * CH) {
    int g = i >> 6;
    out[i] = sums[i] / fmaxf(counts[g], 1.0f);
  }
}

// ---------------------------------------------------------------- launcher
extern "C" void kernel_launch(void* const* d_in, const int* in_sizes, int n_in,
                              void* d_out, int out_size, void* d_ws, size_t ws_size,
                              hipStream_t stream) {
  const float* x    = (const float*)d_in[0];
  const int*   ei   = (const int*)  d_in[1];
  const int*   bat  = (const int*)  d_in[2];
  const float* w1l  = (const float*)d_in[3];
  const float* w1r  = (const float*)d_in[4];
  const float* b1   = (const float*)d_in[5];
  const float* w2l  = (const float*)d_in[6];
  const float* w2r  = (const float*)d_in[7];
  const float* b2   = (const float*)d_in[8];

  long N = in_sizes[0] / CH;
  long E = in_sizes[1] / 2;
  const int* src = ei;
  const int* dst = ei + E;

  // workspace layout
  float* msg  = (float*)d_ws;            // N*64
  float* deg  = msg  + (size_t)N * CH;   // N
  float* h1   = deg  + N;                // N*64
  float* h2   = h1   + (size_t)N * CH;   // N*64
  float* psum = h2   + (size_t)N * CH;   // 64*64
  float* pcnt = psum + NGRAPH * CH;      // 64

  int n_tiles = (int)((N + 15) / 16);

  const int  ZB = 256;
  long zn = N * (CH + 1);
  int  zgrid = (int)((zn + ZB - 1) / ZB); if (zgrid > 131072) zgrid = 131072;
  long sn = E * CH;
  int  sgrid = (int)((sn + ZB - 1) / ZB); if (sgrid > 262144) sgrid = 262144;
  long pn = N * CH;
  int  pgrid = (int)((pn + ZB - 1) / ZB); if (pgrid > 131072) pgrid = 131072;
  int  ggrid = n_tiles < 1024 ? n_tiles : 1024;   // grid-stride; weights staged once/block

  // ---- layer 1
  zero_kernel<<<zgrid, ZB, 0, stream>>>(msg, zn);
  scatter_kernel<<<sgrid, ZB, 0, stream>>>(x, src, dst, msg, deg, E);
  sage_gemm_kernel<<<ggrid, 128, 0, stream>>>(msg, deg, x, w1l, w1r, b1, h1, N, n_tiles);

  // ---- layer 2
  zero_kernel<<<zgrid, ZB, 0, stream>>>(msg, zn);
  scatter_kernel<<<sgrid, ZB, 0, stream>>>(h1, src, dst, msg, deg, E);
  sage_gemm_kernel<<<ggrid, 128, 0, stream>>>(msg, deg, h1, w2l, w2r, b2, h2, N, n_tiles);

  // ---- pool
  zero_kernel<<<((NGRAPH * CH + NGRAPH) + ZB - 1) / ZB, ZB, 0, stream>>>(psum, NGRAPH * CH + NGRAPH);
  pool_accum_kernel<<<pgrid, ZB, 0, stream>>>(h2, bat, psum, pcnt, N);
  pool_final_kernel<<<(NGRAPH * CH + ZB - 1) / ZB, ZB, 0, stream>>>(psum, pcnt, (float*)d_out);
}